// _LBlock_43920335568904
// MI455X (gfx1250) — compile-verified
//
#include <hip/hip_runtime.h>
#include <cstdint>

#define NN 65536
#define EE 262144
#define TT 3
#define DD 128
#define HGn 4
#define BB 32
#define ENP (EE + NN)   // edges per type including self loops

typedef __attribute__((ext_vector_type(2))) float v2f;
typedef __attribute__((ext_vector_type(8))) float v8f;

#define EPI_STORE 0
#define EPI_BLEND 1
#define EPI_ADD   2

__device__ __forceinline__ float leakyf(float x) { return x > 0.f ? x : 0.2f * x; }
__device__ __forceinline__ float geluf(float x) {
  return 0.5f * x * (1.f + erff(x * 0.70710678118654752f));
}
// float atomic max via ordered-int trick (valid for non-NaN values)
__device__ __forceinline__ void atomicMaxF(float* addr, float v) {
  if (v >= 0.f) atomicMax((int*)addr, __float_as_int(v));
  else          atomicMin((unsigned int*)addr, __float_as_uint(v));
}

__global__ void fill_k(float* __restrict__ p, float v, int n) {
  int i = blockIdx.x * blockDim.x + threadIdx.x;
  if (i < n) p[i] = v;
}

// -------- fp32 WMMA GEMM: C[N x 128] = A[N x 128] @ W[128 x 128] (+bias, epilogue) ------
// A-strip staged into LDS with CDNA5 async-to-LDS copies (ASYNCcnt), then consumed
// by v_wmma_f32_16x16x4_f32 with ds_load-fed A fragments.
__global__ __launch_bounds__(256) void gemm128(
    const float* __restrict__ A, const float* __restrict__ W,
    const float* __restrict__ bias, float* __restrict__ C,
    const float* __restrict__ Xold, const float* __restrict__ skipp, int epi)
{
  __shared__ __align__(16) float As[64][132];  // pad 128->132 (row stride 528B, 16B-aligned)
  const int tid = threadIdx.x;
  const int rowBase = blockIdx.x * 64;

  // async stage: 64x128 floats = 2048 x 16B; 256 lanes x 8 issues
  const unsigned ldsBase = (unsigned)(uintptr_t)(&As[0][0]);
#pragma unroll
  for (int i = tid; i < 64 * 32; i += 256) {
    const int r  = i >> 5;
    const int c4 = (i & 31) << 2;
    const unsigned lds = ldsBase + (unsigned)((r * 132 + c4) * 4);
    const int goff = ((rowBase + r) * DD + c4) * 4;   // byte offset (<32MB, fits i32)
    asm volatile("global_load_async_to_lds_b128 %0, %1, %2"
                 :: "v"(lds), "v"(goff), "s"(A) : "memory");
  }
  asm volatile("s_wait_asynccnt 0x0" ::: "memory");
  __syncthreads();

  const int lane  = tid & 31;
  const int wave  = tid >> 5;
  const int laneM = lane & 15;
  const int kHi   = (lane >> 4) << 1;   // lanes 16..31 hold K+2,K+3 of the A/B fragments
  const int col   = wave * 16 + laneM;

  v8f acc[4] = {};
  for (int k0 = 0; k0 < DD; k0 += 4) {
    v2f bf;
    bf.x = W[(k0 + kHi) * DD + col];
    bf.y = W[(k0 + kHi + 1) * DD + col];
#pragma unroll
    for (int rt = 0; rt < 4; ++rt) {
      v2f af;
      af.x = As[rt * 16 + laneM][k0 + kHi];
      af.y = As[rt * 16 + laneM][k0 + kHi + 1];
      acc[rt] = __builtin_amdgcn_wmma_f32_16x16x4_f32(
          false, af, false, bf, (short)0, acc[rt], false, false);
    }
  }
  const float b = bias ? bias[col] : 0.f;
  float sig = 0.f, onem = 0.f;
  if (epi == EPI_BLEND) {
    const float s = *skipp;
    sig = 1.f / (1.f + expf(-s));
    onem = 1.f - sig;
  }
  const int mHi = (lane >> 4) << 3;   // D layout: lanes 16..31 hold rows M+8
#pragma unroll
  for (int rt = 0; rt < 4; ++rt) {
#pragma unroll
    for (int r = 0; r < 8; ++r) {
      const size_t idx = (size_t)(rowBase + rt * 16 + mHi + r) * DD + col;
      float v = acc[rt][r] + b;
      if (epi == EPI_BLEND)    v = sig * v + onem * Xold[idx];
      else if (epi == EPI_ADD) v += C[idx];
      C[idx] = v;
    }
  }
}

// -------- GAT: per-head attention logits from h --------
__global__ void alpha_k(const float* __restrict__ H, const float* __restrict__ as,
                        const float* __restrict__ ad, float* __restrict__ oas,
                        float* __restrict__ oad) {
  int idx = blockIdx.x * blockDim.x + threadIdx.x;
  if (idx >= TT * NN * HGn) return;
  int hd = idx & 3;
  int n  = (idx >> 2) & (NN - 1);
  int t  = idx >> 18;
  const float4* hp = (const float4*)(H + ((size_t)t * NN + n) * DD + hd * 32);
  const float4* ap = (const float4*)(as + (t * HGn + hd) * 32);
  const float4* dp = (const float4*)(ad + (t * HGn + hd) * 32);
  float sa = 0.f, sd = 0.f;
#pragma unroll
  for (int f = 0; f < 8; ++f) {
    float4 h4 = hp[f], a4 = ap[f], d4 = dp[f];
    sa += h4.x * a4.x + h4.y * a4.y + h4.z * a4.z + h4.w * a4.w;
    sd += h4.x * d4.x + h4.y * d4.y + h4.z * d4.z + h4.w * d4.w;
  }
  oas[idx] = sa; oad[idx] = sd;
}

__device__ __forceinline__ void gat_edge(const int* ei, int idx, int& t, int& s, int& d) {
  t = idx / ENP;
  int r = idx - t * ENP;
  if (r < EE) { s = ei[t * 2 * EE + r]; d = ei[t * 2 * EE + EE + r]; }
  else        { s = d = r - EE; }          // self loop
}

__global__ void gat_max_k(const int* __restrict__ ei, const float* __restrict__ asr,
                          const float* __restrict__ ads, float* __restrict__ Mx) {
  int idx = blockIdx.x * blockDim.x + threadIdx.x;
  if (idx >= TT * ENP) return;
  int t, s, d; gat_edge(ei, idx, t, s, d);
  const float* ap = asr + ((size_t)t * NN + s) * 4;
  const float* dp = ads + ((size_t)t * NN + d) * 4;
  float* mp = Mx + ((size_t)t * NN + d) * 4;
#pragma unroll
  for (int h = 0; h < 4; ++h) atomicMaxF(mp + h, leakyf(ap[h] + dp[h]));
}

__global__ void gat_den_k(const int* __restrict__ ei, const float* __restrict__ asr,
                          const float* __restrict__ ads, const float* __restrict__ Mx,
                          float* __restrict__ Den, float* __restrict__ Ex) {
  int idx = blockIdx.x * blockDim.x + threadIdx.x;
  if (idx >= TT * ENP) return;
  int t, s, d; gat_edge(ei, idx, t, s, d);
  const float* ap = asr + ((size_t)t * NN + s) * 4;
  const float* dp = ads + ((size_t)t * NN + d) * 4;
  const float* mp = Mx + ((size_t)t * NN + d) * 4;
  float* dn = Den + ((size_t)t * NN + d) * 4;
#pragma unroll
  for (int h = 0; h < 4; ++h) {
    float ex = expf(leakyf(ap[h] + dp[h]) - mp[h]);
    Ex[(size_t)idx * 4 + h] = ex;
    atomicAdd(dn + h, ex);
  }
}

__global__ void gat_agg_k(const int* __restrict__ ei, const float* __restrict__ Ex,
                          const float* __restrict__ Den, const float* __restrict__ H,
                          float* __restrict__ O) {
  int gid  = blockIdx.x * blockDim.x + threadIdx.x;   // exactly TT*ENP*32 threads
  int c4   = (gid & 31) << 2;                         // 4 channels, same head
  int eidx = gid >> 5;
  int t, s, d; gat_edge(ei, eidx, t, s, d);
  int h = c4 >> 5;
  float w = Ex[(size_t)eidx * 4 + h] / (Den[((size_t)t * NN + d) * 4 + h] + 1e-16f);
  const float4 hv = *(const float4*)(H + ((size_t)t * NN + s) * DD + c4);
  float* op = O + (size_t)d * DD + c4;
  atomicAdd(op + 0, w * hv.x);
  atomicAdd(op + 1, w * hv.y);
  atomicAdd(op + 2, w * hv.z);
  atomicAdd(op + 3, w * hv.w);
}

// -------- residual + rmsnorm + leaky_relu (optional summed per-type bias) --------
__global__ __launch_bounds__(128) void res_k(
    const float* __restrict__ xold, const float* __restrict__ xn,
    const float* __restrict__ b3, int tcnt, const float* __restrict__ scale,
    float* __restrict__ xout) {
  __shared__ float red[128];
  int n = blockIdx.x, c = threadIdx.x;
  float v = xn[(size_t)n * DD + c];
  for (int t = 0; t < tcnt; ++t) v += b3[t * DD + c];
  red[c] = v * v;
  __syncthreads();
  for (int s = 64; s > 0; s >>= 1) { if (c < s) red[c] += red[c + s]; __syncthreads(); }
  float r = rsqrtf(red[0] * (1.f / 128.f) + 1e-6f);
  float o = xold[(size_t)n * DD + c] + scale[c] * v * r;
  xout[(size_t)n * DD + c] = leakyf(o);
}

// -------- HGT: fold per-type head transforms into effective weights/biases --------
__global__ void weff_k(const float* __restrict__ Wk, const float* __restrict__ bk,
                       const float* __restrict__ Wv, const float* __restrict__ bv,
                       const float* __restrict__ a_rel, const float* __restrict__ m_rel,
                       float* __restrict__ We, float* __restrict__ Be) {
  int idx = blockIdx.x * blockDim.x + threadIdx.x;
  if (idx >= 2 * TT * 129 * DD) return;
  int which = idx / (TT * 129 * DD);
  int rem   = idx - which * (TT * 129 * DD);
  int t  = rem / (129 * DD);
  int r2 = rem - t * (129 * DD);
  int i  = r2 / DD;
  int j  = r2 - i * DD;
  int h  = j >> 5, f = j & 31;
  const float* Wsrc = which ? Wv : Wk;
  const float* bsrc = which ? bv : bk;
  const float* rel  = which ? m_rel : a_rel;
  const float* rp = rel + ((size_t)(t * HGn + h) * 32) * 32 + f;   // rel[t][h][d2][f]
  if (i < DD) {
    float sum = 0.f;
    for (int d2 = 0; d2 < 32; ++d2) sum += Wsrc[i * DD + h * 32 + d2] * rp[d2 * 32];
    We[((size_t)which * TT + t) * DD * DD + i * DD + j] = sum;
  } else {
    float sum = 0.f;
    for (int d2 = 0; d2 < 32; ++d2) sum += bsrc[h * 32 + d2] * rp[d2 * 32];
    Be[((size_t)which * TT + t) * DD + j] = sum;
  }
}

__global__ void hgt_score_k(const int* __restrict__ ei, const float* __restrict__ Q,
                            const float* __restrict__ K, const float* __restrict__ prel,
                            float* __restrict__ Sc, float* __restrict__ Mx) {
  int idx = blockIdx.x * blockDim.x + threadIdx.x;
  if (idx >= TT * EE) return;
  int t = idx >> 18, e = idx & (EE - 1);
  int s = ei[t * 2 * EE + e], d = ei[t * 2 * EE + EE + e];
  const float4* qp = (const float4*)(Q + (size_t)d * DD);
  const float4* kp = (const float4*)(K + ((size_t)t * NN + s) * DD);
#pragma unroll
  for (int h = 0; h < 4; ++h) {
    float dot = 0.f;
#pragma unroll
    for (int f = 0; f < 8; ++f) {
      float4 a = qp[h * 8 + f], b = kp[h * 8 + f];
      dot += a.x * b.x + a.y * b.y + a.z * b.z + a.w * b.w;
    }
    float sc = dot * prel[t * 4 + h] * 0.17677669529663687f;  // 1/sqrt(32)
    Sc[(size_t)idx * 4 + h] = sc;
    atomicMaxF(Mx + (size_t)d * 4 + h, sc);
  }
}

__global__ void hgt_den_k(const int* __restrict__ ei, float* __restrict__ Sc,
                          const float* __restrict__ Mx, float* __restrict__ Den) {
  int idx = blockIdx.x * blockDim.x + threadIdx.x;
  if (idx >= TT * EE) return;
  int t = idx >> 18, e = idx & (EE - 1);
  int d = ei[t * 2 * EE + EE + e];
#pragma unroll
  for (int h = 0; h < 4; ++h) {
    float ex = expf(Sc[(size_t)idx * 4 + h] - Mx[(size_t)d * 4 + h]);
    Sc[(size_t)idx * 4 + h] = ex;
    atomicAdd(Den + (size_t)d * 4 + h, ex);
  }
}

__global__ void hgt_agg_k(const int* __restrict__ ei, const float* __restrict__ Ex,
                          const float* __restrict__ V, float* __restrict__ O) {
  int gid = blockIdx.x * blockDim.x + threadIdx.x;   // exactly TT*EE*32 threads
  int c4  = (gid & 31) << 2;
  int fe  = gid >> 5;
  int t = fe >> 18, e = fe & (EE - 1);
  int s = ei[t * 2 * EE + e], d = ei[t * 2 * EE + EE + e];
  float w = Ex[(size_t)fe * 4 + (c4 >> 5)];
  const float4 vv = *(const float4*)(V + ((size_t)t * NN + s) * DD + c4);
  float* op = O + (size_t)d * DD + c4;
  atomicAdd(op + 0, w * vv.x);
  atomicAdd(op + 1, w * vv.y);
  atomicAdd(op + 2, w * vv.z);
  atomicAdd(op + 3, w * vv.w);
}

__global__ void hgt_fin_k(float* __restrict__ O, const float* __restrict__ Den) {
  int idx = blockIdx.x * blockDim.x + threadIdx.x;   // NN*32
  int n = idx >> 5, c4 = (idx & 31) << 2;
  float dn = Den[(size_t)n * 4 + (c4 >> 5)] + 1e-16f;
  float4* p = (float4*)(O + (size_t)n * DD + c4);
  float4 v = *p;
  v.x = geluf(v.x / dn); v.y = geluf(v.y / dn);
  v.z = geluf(v.z / dn); v.w = geluf(v.w / dn);
  *p = v;
}

// -------- FiLM --------
__global__ void film1_k(const float* __restrict__ z, const float* __restrict__ W1,
                        const float* __restrict__ b1, float* __restrict__ Hid) {
  int idx = blockIdx.x * blockDim.x + threadIdx.x;
  if (idx >= BB * 256) return;
  int b = idx >> 8, j = idx & 255;
  float s = b1[j];
  for (int k = 0; k < 128; ++k) s += z[b * 128 + k] * W1[k * 256 + j];
  Hid[idx] = geluf(s);
}

__global__ void film2_k(const float* __restrict__ Hid, const float* __restrict__ W2,
                        const float* __restrict__ b2, float* __restrict__ Gam,
                        float* __restrict__ Bet) {
  int idx = blockIdx.x * blockDim.x + threadIdx.x;
  if (idx >= BB * 256) return;
  int b = idx >> 8, j = idx & 255;
  float s = b2[j];
  for (int k = 0; k < 256; ++k) s += Hid[b * 256 + k] * W2[k * 256 + j];
  float v = 0.1f * tanhf(s);
  if (j < 128) Gam[b * 128 + j] = v;
  else         Bet[b * 128 + (j - 128)] = v;
}

__global__ void final_k(const float* __restrict__ X, const float* __restrict__ Gam,
                        const float* __restrict__ Bet, const int* __restrict__ bszp,
                        float* __restrict__ out) {
  int idx = blockIdx.x * blockDim.x + threadIdx.x;   // NN*32
  int n = idx >> 5, c4 = (idx & 31) << 2;
  int bsz = *bszp;
  int chunk = NN / bsz;
  int bi = n / chunk;
  const float4 g = *(const float4*)(Gam + bi * 128 + c4);
  const float4 bt = *(const float4*)(Bet + bi * 128 + c4);
  const float4 x = *(const float4*)(X + (size_t)n * DD + c4);
  float4 o;
  o.x = (1.f + g.x) * x.x + bt.x;
  o.y = (1.f + g.y) * x.y + bt.y;
  o.z = (1.f + g.z) * x.z + bt.z;
  o.w = (1.f + g.w) * x.w + bt.w;
  *(float4*)(out + (size_t)n * DD + c4) = o;
}

extern "C" void kernel_launch(void* const* d_in, const int* in_sizes, int n_in,
                              void* d_out, int out_size, void* d_ws, size_t ws_size,
                              hipStream_t stream) {
  const float* x_cell = (const float*)d_in[0];
  const float* z_h    = (const float*)d_in[1];
  const float* x_emb  = (const float*)d_in[2];
  const int*   ei     = (const int*)d_in[3];
  const int*   bszp   = (const int*)d_in[4];
  const float* g1W  = (const float*)d_in[5];
  const float* g1as = (const float*)d_in[6];
  const float* g1ad = (const float*)d_in[7];
  const float* g1b  = (const float*)d_in[8];
  const float* g2W  = (const float*)d_in[9];
  const float* g2as = (const float*)d_in[10];
  const float* g2ad = (const float*)d_in[11];
  const float* g2b  = (const float*)d_in[12];
  const float* n1s  = (const float*)d_in[13];
  const float* n2s  = (const float*)d_in[14];
  const float* n3s  = (const float*)d_in[15];
  const float* Wk   = (const float*)d_in[16];
  const float* bk   = (const float*)d_in[17];
  const float* Wq   = (const float*)d_in[18];
  const float* bq   = (const float*)d_in[19];
  const float* Wv   = (const float*)d_in[20];
  const float* bv   = (const float*)d_in[21];
  const float* a_rel= (const float*)d_in[22];
  const float* m_rel= (const float*)d_in[23];
  const float* p_rel= (const float*)d_in[24];
  const float* Wo   = (const float*)d_in[25];
  const float* bo   = (const float*)d_in[26];
  const float* skp  = (const float*)d_in[27];
  const float* injW = (const float*)d_in[28];
  const float* injb = (const float*)d_in[29];
  const float* fW1  = (const float*)d_in[30];
  const float* fb1  = (const float*)d_in[31];
  const float* fW2  = (const float*)d_in[32];
  const float* fb2  = (const float*)d_in[33];
  float* out = (float*)d_out;

  float* ws = (float*)d_ws;
  size_t off = 0;
  float* X0  = ws + off; off += (size_t)NN * DD;
  float* X1  = ws + off; off += (size_t)NN * DD;
  float* O   = ws + off; off += (size_t)NN * DD;
  float* H   = ws + off; off += (size_t)TT * NN * DD;   // gat h / hgt k_t, then v_t
  float* Q   = ws + off; off += (size_t)NN * DD;
  float* Mx  = ws + off; off += (size_t)TT * NN * HGn;
  float* Den = ws + off; off += (size_t)TT * NN * HGn;
  float* Ex  = ws + off; off += (size_t)TT * ENP * HGn;
  float* Asr = ws + off; off += (size_t)TT * NN * HGn;
  float* Ads = ws + off; off += (size_t)TT * NN * HGn;
  float* We  = ws + off; off += (size_t)2 * TT * DD * DD;
  float* Be  = ws + off; off += (size_t)2 * TT * DD;
  float* Hid = ws + off; off += (size_t)BB * 2 * DD;
  float* Gam = ws + off; off += (size_t)BB * DD;
  float* Bet = ws + off; off += (size_t)BB * DD;
  (void)ws_size; (void)in_sizes; (void)n_in; (void)out_size;

  const int TB = 256;
  auto blocks = [](int n, int tb) { return (n + tb - 1) / tb; };

  // ---------------- GAT layer (run twice) ----------------
  auto run_gat = [&](const float* xin, const float* gW, const float* gas,
                     const float* gad, const float* gb, const float* nrm, float* xout) {
    for (int t = 0; t < TT; ++t)
      gemm128<<<NN / 64, 256, 0, stream>>>(xin, gW + (size_t)t * DD * DD, nullptr,
                                           H + (size_t)t * NN * DD, nullptr, nullptr, EPI_STORE);
    fill_k<<<blocks(TT * NN * HGn, TB), TB, 0, stream>>>(Mx, -1e30f, TT * NN * HGn);
    fill_k<<<blocks(TT * NN * HGn, TB), TB, 0, stream>>>(Den, 0.f, TT * NN * HGn);
    fill_k<<<blocks(NN * DD, TB), TB, 0, stream>>>(O, 0.f, NN * DD);
    alpha_k<<<blocks(TT * NN * HGn, TB), TB, 0, stream>>>(H, gas, gad, Asr, Ads);
    gat_max_k<<<blocks(TT * ENP, TB), TB, 0, stream>>>(ei, Asr, Ads, Mx);
    gat_den_k<<<blocks(TT * ENP, TB), TB, 0, stream>>>(ei, Asr, Ads, Mx, Den, Ex);
    gat_agg_k<<<(TT * ENP * 32) / TB, TB, 0, stream>>>(ei, Ex, Den, H, O);
    res_k<<<NN, 128, 0, stream>>>(xin, O, gb, TT, nrm, xout);
  };
  run_gat(x_cell, g1W, g1as, g1ad, g1b, n1s, X0);   // -> x1
  run_gat(X0,     g2W, g2as, g2ad, g2b, n2s, X1);   // -> x2

  // ---------------- HGT layer ----------------
  weff_k<<<blocks(2 * TT * 129 * DD, TB), TB, 0, stream>>>(Wk, bk, Wv, bv, a_rel, m_rel, We, Be);
  gemm128<<<NN / 64, 256, 0, stream>>>(X1, Wq, bq, Q, nullptr, nullptr, EPI_STORE);
  for (int t = 0; t < TT; ++t)        // k_t into H
    gemm128<<<NN / 64, 256, 0, stream>>>(X1, We + (size_t)t * DD * DD, Be + (size_t)t * DD,
                                         H + (size_t)t * NN * DD, nullptr, nullptr, EPI_STORE);
  fill_k<<<blocks(NN * HGn, TB), TB, 0, stream>>>(Mx, -1e30f, NN * HGn);
  fill_k<<<blocks(NN * HGn, TB), TB, 0, stream>>>(Den, 0.f, NN * HGn);
  fill_k<<<blocks(NN * DD, TB), TB, 0, stream>>>(O, 0.f, NN * DD);
  hgt_score_k<<<blocks(TT * EE, TB), TB, 0, stream>>>(ei, Q, H, p_rel, Ex, Mx);
  for (int t = 0; t < TT; ++t)        // v_t overwrites H (k_t now dead)
    gemm128<<<NN / 64, 256, 0, stream>>>(X1, We + (size_t)(TT + t) * DD * DD,
                                         Be + (size_t)(TT + t) * DD,
                                         H + (size_t)t * NN * DD, nullptr, nullptr, EPI_STORE);
  hgt_den_k<<<blocks(TT * EE, TB), TB, 0, stream>>>(ei, Ex, Mx, Den);
  hgt_agg_k<<<(TT * EE * 32) / TB, TB, 0, stream>>>(ei, Ex, H, O);
  hgt_fin_k<<<(NN * 32) / TB, TB, 0, stream>>>(O, Den);
  gemm128<<<NN / 64, 256, 0, stream>>>(O, Wo, bo, X0, X1, skp, EPI_BLEND);  // sig*out+(1-sig)*x2
  res_k<<<NN, 128, 0, stream>>>(X1, X0, nullptr, 0, n3s, X0);               // -> x3
  gemm128<<<NN / 64, 256, 0, stream>>>(x_emb, injW, injb, X0, nullptr, nullptr, EPI_ADD);

  // ---------------- FiLM ----------------
  film1_k<<<blocks(BB * 256, TB), TB, 0, stream>>>(z_h, fW1, fb1, Hid);
  film2_k<<<blocks(BB * 256, TB), TB, 0, stream>>>(Hid, fW2, fb2, Gam, Bet);
  final_k<<<(NN * 32) / TB, TB, 0, stream>>>(X0, Gam, Bet, bszp, out);
}